// Sampler_42296837931299
// MI455X (gfx1250) — compile-verified
//
#include <hip/hip_runtime.h>
#include <stdint.h>

// ---------------- static resampler geometry (16k -> 24k, width 6) ----------
//   in_unit = 2, out_unit = 3, first_indices = {-6,-5,-4}, W = 13 taps
#define NPHASE   3
#define NTAPS    13
#define THREADS  256
#define TU       1024                 // output units per block
#define VU       (TU / THREADS)       // units per thread (4)
#define STAGE_F  (2 * TU + 16)        // floats staged in LDS: x[2u0-8 .. 2u0+2055]

__global__ __launch_bounds__(THREADS)
void resample_poly_16k24k(const float* __restrict__ x,
                          const float* __restrict__ wts,
                          float* __restrict__ out,
                          int L,        // input length per channel
                          int tot,      // output length per channel (== 3*NU)
                          int NU)       // output units per channel
{
    __shared__ float smem[STAGE_F];

    const int tid = threadIdx.x;
    const long long u0 = (long long)blockIdx.x * TU;   // first unit of this tile
    const int c = blockIdx.y;

    const float* __restrict__ xc = x + (long long)c * L;
    float* __restrict__ oc = out + (long long)c * tot;

    // ---- pull the 3x13 filter bank into SGPRs (uniform across the wave) ----
    float wv[NPHASE][NTAPS];
#pragma unroll
    for (int i = 0; i < NPHASE; ++i)
#pragma unroll
        for (int j = 0; j < NTAPS; ++j)
            wv[i][j] = __uint_as_float(
                __builtin_amdgcn_readfirstlane(__float_as_uint(wts[i * NTAPS + j])));

    // ---- async-DMA the input tile into LDS (CDNA5 ASYNCcnt path) ----------
    // smem[idx] mirrors x[gbase + idx], gbase = 2*u0 - 8 (16B aligned).
    const long long gbase = 2 * u0 - 8;
    for (int q4 = tid; q4 < STAGE_F / 4; q4 += THREADS) {
        const long long n0 = gbase + 4LL * q4;
        if (n0 >= 0 && n0 + 4 <= (long long)L) {
            const uint32_t lds_addr = (uint32_t)(uintptr_t)(&smem[4 * q4]);
            const uint64_t gaddr    = (uint64_t)(uintptr_t)(xc + n0);
            asm volatile("global_load_async_to_lds_b128 %0, %1, off"
                         :: "v"(lds_addr), "v"(gaddr)
                         : "memory");
        } else {
            // array edges only (first/last tiles): guarded scalar fill
#pragma unroll
            for (int e = 0; e < 4; ++e) {
                const long long n = n0 + e;
                smem[4 * q4 + e] = (n >= 0 && n < (long long)L) ? xc[n] : 0.0f;
            }
        }
    }
    asm volatile("s_wait_asynccnt 0" ::: "memory");  // drain this wave's DMA
    __syncthreads();                                 // make all waves' LDS visible

    // ---- compute: each thread does VU units at stride THREADS ------------
    // (stride-THREADS unit mapping => stride-2-dword LDS reads, conflict-free)
#pragma unroll
    for (int r = 0; r < VU; ++r) {
        const int ul = tid + r * THREADS;            // unit within tile
        const long long u = u0 + ul;
        if (u < (long long)NU) {
            // window x[2u-6 .. 2u+9] lives at smem[2*ul+2 .. 2*ul+17]
            float xw[16];
            const float2* s2 = (const float2*)(&smem[2 * ul + 2]);   // 8B aligned
#pragma unroll
            for (int h = 0; h < 8; ++h) {
                const float2 v = s2[h];
                xw[2 * h]     = v.x;
                xw[2 * h + 1] = v.y;
            }
            float a0 = 0.0f, a1 = 0.0f, a2 = 0.0f;
#pragma unroll
            for (int j = 0; j < NTAPS; ++j) {
                a0 = fmaf(xw[j],     wv[0][j], a0);   // phase 0: f = -6
                a1 = fmaf(xw[j + 1], wv[1][j], a1);   // phase 1: f = -5
                a2 = fmaf(xw[j + 2], wv[2][j], a2);   // phase 2: f = -4
            }
            // tot == 3*NU, so 3u+2 <= tot-1: unguarded contiguous triple store
            const long long ob = 3 * u;
            oc[ob]     = a0;
            oc[ob + 1] = a1;
            oc[ob + 2] = a2;
        }
    }
}

extern "C" void kernel_launch(void* const* d_in, const int* in_sizes, int n_in,
                              void* d_out, int out_size, void* d_ws, size_t ws_size,
                              hipStream_t stream)
{
    (void)n_in; (void)d_ws; (void)ws_size;
    const float* x = (const float*)d_in[0];   // (8, 4,000,000) f32
    const float* w = (const float*)d_in[1];   // (3, 13) f32
    float* out = (float*)d_out;               // (8, 6,000,000) f32

    const int C   = 8;
    const int L   = in_sizes[0] / C;          // 4,000,000
    const int tot = out_size / C;             // 6,000,000
    const int NU  = tot / NPHASE;             // 2,000,000 output units (exact)
    const int nblk = (NU + TU - 1) / TU;      // 1954 tiles per channel

    dim3 grid(nblk, C, 1);
    resample_poly_16k24k<<<grid, THREADS, 0, stream>>>(x, w, out, L, tot, NU);
}